// GCN_53455162966032
// MI455X (gfx1250) — compile-verified
//
#include <hip/hip_runtime.h>
#include <math.h>

typedef float v2f __attribute__((ext_vector_type(2)));
typedef float v8f __attribute__((ext_vector_type(8)));

#define NEG_SLOPE 0.2f

// ---------------- fp32 WMMA GEMM: C[M,NC] = A[M,K] @ B[K,NC] ----------------
// One wave computes a 16 x (16*NB) strip via V_WMMA_F32_16X16X4_F32 with NB
// accumulators; each A fragment is reused NB times (register blocking along N).
// Layout per CDNA5 ISA 7.12.2:
//   A (16x4):  lane = m + 16*(k>>1), vgpr = k&1
//   B (4x16):  lane = n + 16*(k>>1), vgpr = k&1
//   C/D (16x16): vgpr i holds row (i + 8*(lane>>4)), col = lane&15
template<int K, int NC, int NB>
__global__ void gemm_wmma_f32(const float* __restrict__ A,
                              const float* __restrict__ B,
                              float* __restrict__ C, int M) {
  constexpr int NTB = (NC / 16) / NB;   // N-strip groups per row of tiles
  int wave = blockIdx.x * (blockDim.x >> 5) + (threadIdx.x >> 5);
  int mt = wave / NTB, ng = wave % NTB;
  if (mt * 16 >= M) return;
  int lane = threadIdx.x & 31;
  int l = lane & 15, hi = lane >> 4;
  const float* arow  = A + (size_t)(mt * 16 + l) * K + 2 * hi;
  const float* bbase = B + ng * (NB * 16) + l + (size_t)(2 * hi) * NC;

  v8f acc[NB];
#pragma unroll
  for (int t = 0; t < NB; ++t) acc[t] = (v8f){0.f,0.f,0.f,0.f,0.f,0.f,0.f,0.f};

  for (int k0 = 0; k0 < K; k0 += 4) {
    v2f a;
    a.x = arow[k0];
    a.y = arow[k0 + 1];
    const float* bk = bbase + (size_t)k0 * NC;
#pragma unroll
    for (int t = 0; t < NB; ++t) {
      v2f b;
      b.x = bk[t * 16];
      b.y = bk[t * 16 + NC];
      acc[t] = __builtin_amdgcn_wmma_f32_16x16x4_f32(
          /*neg_a=*/false, a, /*neg_b=*/false, b,
          /*c_mod=*/(short)0, acc[t], /*reuse_a=*/false, /*reuse_b=*/false);
    }
  }

  float* crow = C + (size_t)(mt * 16 + 8 * hi) * NC + ng * (NB * 16) + l;
#pragma unroll
  for (int t = 0; t < NB; ++t)
#pragma unroll
    for (int i = 0; i < 8; ++i) crow[(size_t)i * NC + t * 16] = acc[t][i];
}

// ---------------- helpers ----------------
__device__ __forceinline__ void edge_sd(const int* __restrict__ ei, int E, long e,
                                        int& s, int& d) {
  if (e < E) { s = ei[e]; d = ei[E + e]; }
  else       { s = (int)(e - E); d = s; }   // implicit self-loop
}

__device__ __forceinline__ float leaky(float x) {
  return x > 0.f ? x : NEG_SLOPE * x;
}

// float atomic-max via signed-int max / unsigned-int min (init must be -inf)
__device__ __forceinline__ void atomicMaxF(float* a, float v) {
  if (v >= 0.f) atomicMax((int*)a, __float_as_int(v));
  else          atomicMin((unsigned int*)a, __float_as_uint(v));
}

__global__ void fill_f32(float* __restrict__ p, float v, long n) {
  long i = (long)blockIdx.x * blockDim.x + threadIdx.x;
  if (i < n) p[i] = v;
}

// per-node attention coefficients + init segmax/denom. h: [N, H*F]
__global__ void gat_alpha(const float* __restrict__ h,
                          const float* __restrict__ a_src,
                          const float* __restrict__ a_dst,
                          float* __restrict__ as, float* __restrict__ ad,
                          float* __restrict__ sm, float* __restrict__ dn,
                          int N, int H, int F) {
  int idx = blockIdx.x * blockDim.x + threadIdx.x;  // idx = n*H + head
  if (idx >= N * H) return;
  int hd = idx % H;
  const float* hp = h + (size_t)idx * F;
  float s1 = 0.f, s2 = 0.f;
  for (int f = 0; f < F; ++f) {
    float v = hp[f];
    s1 += v * a_src[hd * F + f];
    s2 += v * a_dst[hd * F + f];
  }
  as[idx] = s1; ad[idx] = s2;
  sm[idx] = -INFINITY; dn[idx] = 0.f;
}

// pass 1: segment max of leaky logits (+ optional degree count for GCN)
__global__ void edge_max(const int* __restrict__ ei, int E, int N, int H,
                         const float* __restrict__ as, const float* __restrict__ ad,
                         float* __restrict__ sm, float* __restrict__ deg) {
  long e = (long)blockIdx.x * blockDim.x + threadIdx.x;
  if (e >= (long)E + N) return;
  int s, d; edge_sd(ei, E, e, s, d);
  for (int h = 0; h < H; ++h) {
    float lg = leaky(as[(size_t)s * H + h] + ad[(size_t)d * H + h]);
    atomicMaxF(&sm[(size_t)d * H + h], lg);
  }
  if (deg) atomicAdd(&deg[d], 1.f);
}

// pass 2: segment sum of exp(logit - segmax)
__global__ void edge_denom(const int* __restrict__ ei, int E, int N, int H,
                           const float* __restrict__ as, const float* __restrict__ ad,
                           const float* __restrict__ sm, float* __restrict__ dn) {
  long e = (long)blockIdx.x * blockDim.x + threadIdx.x;
  if (e >= (long)E + N) return;
  int s, d; edge_sd(ei, E, e, s, d);
  for (int h = 0; h < H; ++h) {
    float lg = leaky(as[(size_t)s * H + h] + ad[(size_t)d * H + h]);
    atomicAdd(&dn[(size_t)d * H + h], __expf(lg - sm[(size_t)d * H + h]));
  }
}

// pass 3: weighted message scatter-add. One thread per (edge, 4 features).
// Recomputes alpha from node-level arrays (all L2-resident) instead of
// materializing [E,H] tensors.
__global__ void edge_msg(const int* __restrict__ ei, int E, int N, int H, int F,
                         const float* __restrict__ as, const float* __restrict__ ad,
                         const float* __restrict__ sm, const float* __restrict__ dn,
                         const float* __restrict__ hfeat, float* __restrict__ agg) {
  const int C = H * F, vpe = C / 4;
  long idx = (long)blockIdx.x * blockDim.x + threadIdx.x;
  if (idx >= ((long)E + N) * vpe) return;
  long e = idx / vpe;
  int q = (int)(idx % vpe);
  int h = (q * 4) / F;
  int s, d; edge_sd(ei, E, e, s, d);
  float lg = leaky(as[(size_t)s * H + h] + ad[(size_t)d * H + h]);
  float coeff = __expf(lg - sm[(size_t)d * H + h]) / dn[(size_t)d * H + h];
  const float4 v = *((const float4*)(hfeat + (size_t)s * C) + q);
  float* ap = agg + (size_t)d * C + q * 4;
  atomicAdd(ap + 0, v.x * coeff);
  atomicAdd(ap + 1, v.y * coeff);
  atomicAdd(ap + 2, v.z * coeff);
  atomicAdd(ap + 3, v.w * coeff);
}

__global__ void elu_bias(const float* __restrict__ agg, const float* __restrict__ b,
                         float* __restrict__ out, long n, int C) {
  long i = (long)blockIdx.x * blockDim.x + threadIdx.x;
  if (i >= n) return;
  float v = agg[i] + b[i % C];
  out[i] = v > 0.f ? v : __expf(v) - 1.f;   // ELU
}

__global__ void deg_to_dinv(float* __restrict__ deg, int N) {
  int i = blockIdx.x * blockDim.x + threadIdx.x;
  if (i >= N) return;
  float d = deg[i];
  deg[i] = d > 0.f ? rsqrtf(d) : 0.f;
}

__global__ void bias_init(float* __restrict__ out, const float* __restrict__ b,
                          long n, int C) {
  long i = (long)blockIdx.x * blockDim.x + threadIdx.x;
  if (i < n) out[i] = b[i % C];
}

// GCN scatter: out[d,c] += h[s,c] * dinv[s] * dinv[d]
__global__ void gcn_edge(const int* __restrict__ ei, int E, int N, int C,
                         const float* __restrict__ dinv,
                         const float* __restrict__ h, float* __restrict__ out) {
  long idx = (long)blockIdx.x * blockDim.x + threadIdx.x;
  if (idx >= ((long)E + N) * C) return;
  long e = idx / C;
  int c = (int)(idx % C);
  int s, d; edge_sd(ei, E, e, s, d);
  float norm = dinv[s] * dinv[d];
  atomicAdd(&out[(size_t)d * C + c], h[(size_t)s * C + c] * norm);
}

// ---------------- host ----------------
static inline int nb(long n, int t = 256) { return (int)((n + t - 1) / t); }

extern "C" void kernel_launch(void* const* d_in, const int* in_sizes, int n_in,
                              void* d_out, int out_size, void* d_ws, size_t ws_size,
                              hipStream_t stream) {
  const float* x      = (const float*)d_in[0];
  const int*   ei     = (const int*)  d_in[1];
  const float* W1     = (const float*)d_in[2];
  const float* a_src1 = (const float*)d_in[3];
  const float* a_dst1 = (const float*)d_in[4];
  const float* b1     = (const float*)d_in[5];
  const float* W2     = (const float*)d_in[6];
  const float* a_src2 = (const float*)d_in[7];
  const float* a_dst2 = (const float*)d_in[8];
  const float* b2     = (const float*)d_in[9];
  const float* W3     = (const float*)d_in[10];
  const float* b3     = (const float*)d_in[11];

  const int IN = 128, H1 = 8, F1 = 32, C1 = H1 * F1;  // 256
  const int C2 = 16, NC = 16;
  const int N = in_sizes[0] / IN;
  const int E = in_sizes[1] / 2;
  const long EA = (long)E + N;

  // workspace carve-out (~120 MB for N=50000)
  float* p      = (float*)d_ws;
  float* bufA   = p;                  p += (size_t)N * C1;  // h1, later out1
  float* agg1   = p;                  p += (size_t)N * C1;
  float* as1    = p;                  p += (size_t)N * H1;
  float* ad1    = p;                  p += (size_t)N * H1;
  float* sm1    = p;                  p += (size_t)N * H1;
  float* dn1    = p;                  p += (size_t)N * H1;
  float* bufB   = p;                  p += (size_t)N * C2;  // h2, later out2
  float* agg2   = p;                  p += (size_t)N * C2;
  float* as2    = p;                  p += (size_t)N;
  float* ad2    = p;                  p += (size_t)N;
  float* sm2    = p;                  p += (size_t)N;
  float* dn2    = p;                  p += (size_t)N;
  float* deg    = p;                  p += (size_t)N;       // deg -> dinv
  float* h3     = p;                  p += (size_t)N * NC;
  (void)ws_size; (void)n_in; (void)out_size;

  float* out = (float*)d_out;
  const int T = 256;

  // ---- GAT layer 1 ----
  {
    // NB=4: each wave computes a 16x64 strip; A fragments reused 4x.
    long waves = ((long)(N + 15) / 16) * ((C1 / 16) / 4);
    gemm_wmma_f32<IN, C1, 4><<<nb(waves * 32), T, 0, stream>>>(x, W1, bufA, N);
  }
  fill_f32<<<nb((long)N * C1), T, 0, stream>>>(agg1, 0.f, (long)N * C1);
  fill_f32<<<nb(N), T, 0, stream>>>(deg, 0.f, N);
  gat_alpha<<<nb((long)N * H1), T, 0, stream>>>(bufA, a_src1, a_dst1,
                                                as1, ad1, sm1, dn1, N, H1, F1);
  edge_max<<<nb(EA), T, 0, stream>>>(ei, E, N, H1, as1, ad1, sm1, deg);
  edge_denom<<<nb(EA), T, 0, stream>>>(ei, E, N, H1, as1, ad1, sm1, dn1);
  edge_msg<<<nb(EA * (C1 / 4)), T, 0, stream>>>(ei, E, N, H1, F1,
                                                as1, ad1, sm1, dn1, bufA, agg1);
  elu_bias<<<nb((long)N * C1), T, 0, stream>>>(agg1, b1, bufA, (long)N * C1, C1);
  // bufA now holds elu(out1)

  // ---- GAT layer 2 (1 head, 16 features) ----
  {
    long waves = ((long)(N + 15) / 16) * (C2 / 16);
    gemm_wmma_f32<C1, C2, 1><<<nb(waves * 32), T, 0, stream>>>(bufA, W2, bufB, N);
  }
  fill_f32<<<nb((long)N * C2), T, 0, stream>>>(agg2, 0.f, (long)N * C2);
  gat_alpha<<<nb(N), T, 0, stream>>>(bufB, a_src2, a_dst2,
                                     as2, ad2, sm2, dn2, N, 1, C2);
  edge_max<<<nb(EA), T, 0, stream>>>(ei, E, N, 1, as2, ad2, sm2, nullptr);
  edge_denom<<<nb(EA), T, 0, stream>>>(ei, E, N, 1, as2, ad2, sm2, dn2);
  edge_msg<<<nb(EA * (C2 / 4)), T, 0, stream>>>(ei, E, N, 1, C2,
                                                as2, ad2, sm2, dn2, bufB, agg2);
  elu_bias<<<nb((long)N * C2), T, 0, stream>>>(agg2, b2, bufB, (long)N * C2, C2);
  // bufB now holds elu(out2)

  // ---- GCN layer ----
  deg_to_dinv<<<nb(N), T, 0, stream>>>(deg, N);
  {
    long waves = ((long)(N + 15) / 16) * (NC / 16);
    gemm_wmma_f32<C2, NC, 1><<<nb(waves * 32), T, 0, stream>>>(bufB, W3, h3, N);
  }
  bias_init<<<nb((long)N * NC), T, 0, stream>>>(out, b3, (long)N * NC, NC);
  gcn_edge<<<nb(EA * NC), T, 0, stream>>>(ei, E, N, NC, deg, h3, out);
}